// WaveKANLinear_51728586113695
// MI455X (gfx1250) — compile-verified
//
#include <hip/hip_runtime.h>
#include <hip/hip_bf16.h>

typedef __attribute__((ext_vector_type(2))) float v2f;
typedef __attribute__((ext_vector_type(4))) float v4f;
typedef __attribute__((ext_vector_type(8))) float v8f;

#define TILE_N   32
#define DD       128
#define OO       128
#define OCHUNK   32
#define PSTRIDE  33    // OCHUNK + 1 : conflict-free transpose write + lane-over-o read
#define XSTRIDE  132   // DD + 4    : conflict-free WMMA A-operand v2f reads
#define BSTRIDE  129   // OO + 1

// 2 / (sqrt(3) * pi^0.25)
#define MEX_C 0.8673250705840776f

__device__ __forceinline__ float fast_rcp(float v) {
    return __builtin_amdgcn_rcpf(v);           // v_rcp_f32, ~1 ulp
}

__global__ __launch_bounds__(256)
void wavekan_fused_kernel(const float* __restrict__ x,
                          const float* __restrict__ scale,
                          const float* __restrict__ translation,
                          const float* __restrict__ wav_w,
                          const float* __restrict__ base_w,
                          const float* __restrict__ ln_gamma,
                          const float* __restrict__ ln_beta,
                          float* __restrict__ out)
{
    extern __shared__ float smem[];
    float* xnorm = smem;                       // TILE_N * DD        = 4096 floats
    float* xraw  = xnorm + TILE_N * DD;        // TILE_N * XSTRIDE   = 4224
    float* btile = xraw  + TILE_N * XSTRIDE;   // TILE_N * BSTRIDE   = 4128
    float* pinv  = btile + TILE_N * BSTRIDE;   // DD * PSTRIDE       = 4224
    float* pc    = pinv  + DD * PSTRIDE;       // DD * PSTRIDE
    float* pww   = pc    + DD * PSTRIDE;       // DD * PSTRIDE

    const int tid  = threadIdx.x;
    const int wave = tid >> 5;
    const int lane = tid & 31;
    const int n0   = blockIdx.x * TILE_N;

    // ---------------- Phase 1: LayerNorm, stage x_norm + x_raw ----------------
    {
        v4f g  = *(const v4f*)(ln_gamma + lane * 4);
        v4f be = *(const v4f*)(ln_beta  + lane * 4);
        #pragma unroll
        for (int i = 0; i < 4; ++i) {
            const int r = wave * 4 + i;
            v4f xv = *(const v4f*)(x + (size_t)(n0 + r) * DD + lane * 4);
            float s  = xv.x + xv.y + xv.z + xv.w;
            float s2 = xv.x*xv.x + xv.y*xv.y + xv.z*xv.z + xv.w*xv.w;
            #pragma unroll
            for (int off = 16; off >= 1; off >>= 1) {
                s  += __shfl_xor(s,  off, 32);
                s2 += __shfl_xor(s2, off, 32);
            }
            const float mean = s * (1.0f / DD);
            const float var  = s2 * (1.0f / DD) - mean * mean;
            const float rstd = rsqrtf(var + 1e-5f);
            v4f xn;
            xn.x = (xv.x - mean) * rstd * g.x + be.x;
            xn.y = (xv.y - mean) * rstd * g.y + be.y;
            xn.z = (xv.z - mean) * rstd * g.z + be.z;
            xn.w = (xv.w - mean) * rstd * g.w + be.w;
            *(v4f*)(xnorm + r * DD      + lane * 4) = xn;
            *(v4f*)(xraw  + r * XSTRIDE + lane * 4) = xv;
        }
    }
    __syncthreads();

    // ---------------- Phase 2: base_out = x @ base_w^T via f32 WMMA, SiLU -----
    // 16 tiles of 16x16 over (2 row-tiles x 8 o-tiles); wave w does tiles w, w+8.
    {
        const int half = lane >> 4;
        const int l    = lane & 15;
        #pragma unroll
        for (int p = 0; p < 2; ++p) {
            const int t    = wave + p * 8;
            const int row0 = (t >> 3) * 16;
            const int o0   = (t & 7) * 16;
            v8f acc = {};
            #pragma unroll 8
            for (int k = 0; k < 32; ++k) {
                // A: 16x4 f32 chunk of raw x (lanes 0-15: K=4k,4k+1 ; 16-31: K=4k+2,4k+3)
                v2f a = *(const v2f*)(xraw + (row0 + l) * XSTRIDE + 4 * k + 2 * half);
                // B: 4x16 f32 chunk of base_w^T (rows K, cols o) read from o-major base_w
                v2f b = *(const v2f*)(base_w + (size_t)(o0 + l) * DD + 4 * k + 2 * half);
                acc = __builtin_amdgcn_wmma_f32_16x16x4_f32(
                        false, a, false, b, (short)0, acc, false, false);
            }
            #pragma unroll
            for (int j = 0; j < 8; ++j) {
                const int M = j + half * 8;
                const float v  = acc[j];
                // SiLU via v_rcp_f32 (avoid IEEE div_scale/div_fixup sequence)
                const float sv = v * fast_rcp(1.0f + __expf(-v));
                btile[(row0 + M) * BSTRIDE + o0 + l] = sv;
            }
        }
    }

    // ---------------- Phase 3: wavelet path, 4 o-chunks of 32 -----------------
    const int rbase = wave * 4;
    for (int cc = 0; cc < 4; ++cc) {
        __syncthreads();   // btile ready (first iter) / previous chunk compute done
        const int ob = cc * OCHUNK;

        // Stage params transposed [d][o] with softplus folded:
        //   z = fma(x_norm, inv_s, c)   with c = -t * inv_s,  wwc = MEX_C * w
        #pragma unroll 4
        for (int it = 0; it < 16; ++it) {
            const int e  = tid + it * 256;
            const int oo = e >> 7;
            const int d  = e & 127;
            const size_t gi = (size_t)(ob + oo) * DD + d;
            const float sc  = scale[gi];
            // softplus(sc) = max(sc,0) + log(1 + exp(-|sc|))  (fast exp/log paths)
            const float sp  = fmaxf(sc, 0.0f) + __logf(1.0f + __expf(-fabsf(sc)));
            const float inv = fast_rcp(sp + 0.1f);      // s >= 0.1, rcp is safe
            pinv[d * PSTRIDE + oo] = inv;
            pc  [d * PSTRIDE + oo] = -translation[gi] * inv;
            pww [d * PSTRIDE + oo] = MEX_C * wav_w[gi];
        }
        __syncthreads();

        // Each wave: rows rbase..rbase+3, lanes own output o = ob + lane.
        float a0 = 0.f, a1 = 0.f, a2 = 0.f, a3 = 0.f;
        #pragma unroll 4
        for (int d = 0; d < DD; ++d) {
            const float inv = pinv[d * PSTRIDE + lane];
            const float c   = pc  [d * PSTRIDE + lane];
            const float wv  = pww [d * PSTRIDE + lane];
            const float x0 = xnorm[(rbase + 0) * DD + d];   // wave-uniform broadcasts
            const float x1 = xnorm[(rbase + 1) * DD + d];
            const float x2 = xnorm[(rbase + 2) * DD + d];
            const float x3 = xnorm[(rbase + 3) * DD + d];
            // mexican hat (clip dropped: |mex| <= 0.867 < 10 always)
            {   float z = fmaf(x0, inv, c); float z2 = z * z;
                a0 = fmaf((z2 - 1.0f) * __expf(-0.5f * z2), wv, a0); }
            {   float z = fmaf(x1, inv, c); float z2 = z * z;
                a1 = fmaf((z2 - 1.0f) * __expf(-0.5f * z2), wv, a1); }
            {   float z = fmaf(x2, inv, c); float z2 = z * z;
                a2 = fmaf((z2 - 1.0f) * __expf(-0.5f * z2), wv, a2); }
            {   float z = fmaf(x3, inv, c); float z2 = z * z;
                a3 = fmaf((z2 - 1.0f) * __expf(-0.5f * z2), wv, a3); }
        }

        // out = wavelet + silu(base); 128B coalesced stores per row
        out[(size_t)(n0 + rbase + 0) * OO + ob + lane] = a0 + btile[(rbase + 0) * BSTRIDE + ob + lane];
        out[(size_t)(n0 + rbase + 1) * OO + ob + lane] = a1 + btile[(rbase + 1) * BSTRIDE + ob + lane];
        out[(size_t)(n0 + rbase + 2) * OO + ob + lane] = a2 + btile[(rbase + 2) * BSTRIDE + ob + lane];
        out[(size_t)(n0 + rbase + 3) * OO + ob + lane] = a3 + btile[(rbase + 3) * BSTRIDE + ob + lane];
    }
}

extern "C" void kernel_launch(void* const* d_in, const int* in_sizes, int n_in,
                              void* d_out, int out_size, void* d_ws, size_t ws_size,
                              hipStream_t stream) {
    const float* x     = (const float*)d_in[0];
    const float* scale = (const float*)d_in[1];
    const float* trans = (const float*)d_in[2];
    const float* wav_w = (const float*)d_in[3];
    const float* basew = (const float*)d_in[4];
    const float* gam   = (const float*)d_in[5];
    const float* bet   = (const float*)d_in[6];
    float* out = (float*)d_out;

    const int N = in_sizes[0] / DD;          // B*T = 8192
    const int grid = N / TILE_N;             // 256 blocks
    const size_t smem_bytes =
        (size_t)(TILE_N * DD + TILE_N * XSTRIDE + TILE_N * BSTRIDE + 3 * DD * PSTRIDE)
        * sizeof(float);                     // 100480 B

    wavekan_fused_kernel<<<grid, 256, smem_bytes, stream>>>(
        x, scale, trans, wav_w, basew, gam, bet, out);
}